// VPN_14585708937910
// MI455X (gfx1250) — compile-verified
//
#include <hip/hip_runtime.h>
#include <math.h>

// ---------------------------------------------------------------------------
// Problem constants (match the reference)
// ---------------------------------------------------------------------------
#define NN 4096   // nodes
#define FF 256    // feature dim
#define EE 65536  // edges

typedef __attribute__((ext_vector_type(2))) float v2f;
typedef __attribute__((ext_vector_type(8))) float v8f;

// V_WMMA_F32_16X16X4_F32 : D(16x16 f32) = A(16x4 f32) * B(4x16 f32) + C
// 8-arg pattern: (neg_a, A, neg_b, B, c_mod, C, reuse_a, reuse_b)
static __device__ __forceinline__ v8f wmma4(v2f a, v2f b, v8f c) {
  return __builtin_amdgcn_wmma_f32_16x16x4_f32(false, a, false, b, (short)0, c,
                                               false, false);
}

// ---------------------------------------------------------------------------
// 1. Adjacency construction
// ---------------------------------------------------------------------------
__global__ void k_zero(float* __restrict__ p, size_t n) {
  size_t i = (size_t)blockIdx.x * blockDim.x + threadIdx.x;
  if (i < n) p[i] = 0.0f;
}

__global__ void k_scatter(const int* __restrict__ ei,
                          const float* __restrict__ w, float* __restrict__ A) {
  int e = blockIdx.x * blockDim.x + threadIdx.x;
  if (e < EE) {
    int u = ei[e];
    int v = ei[EE + e];
    atomicAdd(&A[(size_t)u * NN + v], w[e]);
  }
}

__global__ void k_finishA(float* __restrict__ A) {
  size_t i = (size_t)blockIdx.x * blockDim.x + threadIdx.x;
  if (i < (size_t)NN * NN) {
    float v = A[i];
    if ((i >> 12) == (i & (NN - 1))) v += 1.0f;  // + I
    A[i] = v > 1.0f ? 1.0f : v;                  // clamp
  }
}

// ---------------------------------------------------------------------------
// 2. f32 WMMA GEMMs.
//    Fragment layouts per CDNA5 ISA 7.12.2 (wave32):
//      A 16x4 f32 : lane r=lane&15 holds row M=r, K = {kk, kk+1}, kk=2*(lane>>4)
//      B 4x16 f32 : VGPR0 = row K=kk cols 0..15, VGPR1 = row K=kk+1
//      C/D 16x16  : VGPR j, lanes0-15 -> M=j ; lanes16-31 -> M=j+8
//    Each wave: 32x32 output (2x2 WMMA tiles). 8 waves/block -> 128x64 tile.
// ---------------------------------------------------------------------------
__global__ void __launch_bounds__(256) k_gemm_nn(const float* __restrict__ A,
                                                 const float* __restrict__ B,
                                                 float* __restrict__ C, int K) {
  const int lane = threadIdx.x & 31;
  const int wave = threadIdx.x >> 5;
  const int r = lane & 15;
  const int kk = (lane >> 4) << 1;
  const int m0 = blockIdx.y * 128 + (wave >> 1) * 32;
  const int n0 = blockIdx.x * 64 + (wave & 1) * 32;

  v8f c00 = {}, c01 = {}, c10 = {}, c11 = {};
  const float* a0p = A + (size_t)(m0 + r) * NN;
  const float* a1p = A + (size_t)(m0 + 16 + r) * NN;

  for (int k = 0; k < K; k += 4) {
    v2f a0 = *(const v2f*)(a0p + k + kk);
    v2f a1 = *(const v2f*)(a1p + k + kk);
    const float* br0 = B + (size_t)(k + kk) * NN;
    const float* br1 = br0 + NN;
    v2f b0, b1;
    b0.x = br0[n0 + r];
    b0.y = br1[n0 + r];
    b1.x = br0[n0 + 16 + r];
    b1.y = br1[n0 + 16 + r];
    c00 = wmma4(a0, b0, c00);
    c01 = wmma4(a0, b1, c01);
    c10 = wmma4(a1, b0, c10);
    c11 = wmma4(a1, b1, c11);
  }

  const int rowb = m0 + ((lane >> 4) << 3);
#pragma unroll
  for (int j = 0; j < 8; ++j) {
    C[(size_t)(rowb + j) * NN + n0 + r] = c00[j];
    C[(size_t)(rowb + j) * NN + n0 + 16 + r] = c01[j];
    C[(size_t)(rowb + 16 + j) * NN + n0 + r] = c10[j];
    C[(size_t)(rowb + 16 + j) * NN + n0 + 16 + r] = c11[j];
  }
}

// G = X * X^T  (X row-major NN x FF); B fragment loads are contiguous float2
__global__ void __launch_bounds__(256) k_gemm_nt(const float* __restrict__ X,
                                                 float* __restrict__ C) {
  const int lane = threadIdx.x & 31;
  const int wave = threadIdx.x >> 5;
  const int r = lane & 15;
  const int kk = (lane >> 4) << 1;
  const int m0 = blockIdx.y * 128 + (wave >> 1) * 32;
  const int n0 = blockIdx.x * 64 + (wave & 1) * 32;

  v8f c00 = {}, c01 = {}, c10 = {}, c11 = {};
  const float* a0p = X + (size_t)(m0 + r) * FF;
  const float* a1p = X + (size_t)(m0 + 16 + r) * FF;
  const float* b0p = X + (size_t)(n0 + r) * FF;
  const float* b1p = X + (size_t)(n0 + 16 + r) * FF;

  for (int k = 0; k < FF; k += 4) {
    v2f a0 = *(const v2f*)(a0p + k + kk);
    v2f a1 = *(const v2f*)(a1p + k + kk);
    v2f b0 = *(const v2f*)(b0p + k + kk);
    v2f b1 = *(const v2f*)(b1p + k + kk);
    c00 = wmma4(a0, b0, c00);
    c01 = wmma4(a0, b1, c01);
    c10 = wmma4(a1, b0, c10);
    c11 = wmma4(a1, b1, c11);
  }

  const int rowb = m0 + ((lane >> 4) << 3);
#pragma unroll
  for (int j = 0; j < 8; ++j) {
    C[(size_t)(rowb + j) * NN + n0 + r] = c00[j];
    C[(size_t)(rowb + j) * NN + n0 + 16 + r] = c01[j];
    C[(size_t)(rowb + 16 + j) * NN + n0 + r] = c10[j];
    C[(size_t)(rowb + 16 + j) * NN + n0 + 16 + r] = c11[j];
  }
}

// ---------------------------------------------------------------------------
// 3. Post-GEMM scalar passes
// ---------------------------------------------------------------------------
__global__ void k_zerodiag(float* __restrict__ A, float* __restrict__ A2) {
  int i = blockIdx.x * blockDim.x + threadIdx.x;
  if (i < NN) {
    A[(size_t)i * NN + i] = 0.0f;
    A2[(size_t)i * NN + i] = 0.0f;
  }
}

__global__ void __launch_bounds__(256) k_degrees(const float* __restrict__ A,
                                                 const float* __restrict__ A2,
                                                 int* __restrict__ D,
                                                 int* __restrict__ Dr) {
  __shared__ int s1[256], s2[256];
  const int u = blockIdx.x;
  const float* ar = A + (size_t)u * NN;
  const float* a2r = A2 + (size_t)u * NN;
  int c1 = 0, c2 = 0;
  for (int w = threadIdx.x; w < NN; w += 256) {
    c1 += (ar[w] != 0.0f);
    c2 += (a2r[w] != 0.0f);
  }
  s1[threadIdx.x] = c1;
  s2[threadIdx.x] = c2;
  __syncthreads();
  for (int st = 128; st > 0; st >>= 1) {
    if (threadIdx.x < st) {
      s1[threadIdx.x] += s1[threadIdx.x + st];
      s2[threadIdx.x] += s2[threadIdx.x + st];
    }
    __syncthreads();
  }
  if (threadIdx.x == 0) {
    D[u] = s1[0];
    Dr[u] = s2[0];
  }
}

__global__ void k_gather(const int* __restrict__ ei,
                         const float* __restrict__ A,
                         const float* __restrict__ A2,
                         float* __restrict__ auv, float* __restrict__ a2uv) {
  int e = blockIdx.x * blockDim.x + threadIdx.x;
  if (e < EE) {
    int u = ei[e];
    int v = ei[EE + e];
    auv[e] = A[(size_t)u * NN + v];
    a2uv[e] = A2[(size_t)u * NN + v];
  }
}

__global__ void k_sq(const float* __restrict__ X, float* __restrict__ sq) {
  int u = blockIdx.x * blockDim.x + threadIdx.x;
  if (u < NN) {
    const float* x = X + (size_t)u * FF;
    float s = 0.0f;
    for (int f = 0; f < FF; ++f) s += x[f] * x[f];
    sq[u] = s;
  }
}

__global__ void __launch_bounds__(256) k_stats(const int* __restrict__ D,
                                               float* __restrict__ thres) {
  __shared__ float s[256];
  __shared__ float meanv;
  const int t = threadIdx.x;
  float acc = 0.0f;
  for (int i = t; i < NN; i += 256) acc += (float)D[i];
  s[t] = acc;
  __syncthreads();
  for (int st = 128; st > 0; st >>= 1) {
    if (t < st) s[t] += s[t + st];
    __syncthreads();
  }
  if (t == 0) meanv = s[0] / (float)NN;
  __syncthreads();
  float m = meanv;
  acc = 0.0f;
  for (int i = t; i < NN; i += 256) {
    float d = (float)D[i] - m;
    acc += d * d;
  }
  s[t] = acc;
  __syncthreads();
  for (int st = 128; st > 0; st >>= 1) {
    if (t < st) s[t] += s[t + st];
    __syncthreads();
  }
  if (t == 0) thres[0] = m + 2.0f * sqrtf(s[0] / (float)NN);
}

// ---------------------------------------------------------------------------
// 4. Per-edge output. One block (256 threads) per edge. The reference's full
//    NxN argsort-rank collapses to a per-edge count of farther neighbors.
// ---------------------------------------------------------------------------
__global__ void __launch_bounds__(256)
k_edges(const int* __restrict__ ei, const float* __restrict__ w,
        const float* __restrict__ th1, const float* __restrict__ th2,
        const float* __restrict__ A2, const float* __restrict__ G,
        const float* __restrict__ sq, const int* __restrict__ D,
        const int* __restrict__ Dr, const float* __restrict__ auv,
        const float* __restrict__ a2uv, const float* __restrict__ thres,
        float* __restrict__ out) {
  __shared__ int s[256];
  const int e = blockIdx.x;
  const int u = ei[e];
  const int v = ei[EE + e];
  const float a = auv[e];
  const float a2 = a2uv[e];
  const int Du = D[u];
  const int Dru = Dr[u];

  float adj;
  bool needRank = false;
  if ((float)Du > thres[0]) {
    adj = (a == 0.0f) ? 0.0f : (a2 - a);  // highD: zero non-1-hop entries
  } else if (2 * Du > Dru) {
    adj = a2 - a;  // nosparse branch: Z = false
  } else if (a2 == 0.0f) {
    adj = a2 - a;  // not a 2-hop neighbor: Z = false
  } else {
    adj = 0.0f;
    needRank = true;
  }

  if (needRank) {
    const int nz = Dru - 2 * Du;
    const int nzeff = (nz == 0) ? NN : nz;
    const float* a2r = A2 + (size_t)u * NN;
    const float* gr = G + (size_t)u * NN;
    const float squ = sq[u];
    const float dv = squ + sq[v] - 2.0f * gr[v];
    int cnt = 0;
    for (int x = threadIdx.x; x < NN; x += 256) {
      if (a2r[x] != 0.0f) {
        float dx = squ + sq[x] - 2.0f * gr[x];
        cnt += (dx > dv) || ((dx == dv) && (x < v));
      }
    }
    s[threadIdx.x] = cnt;
    __syncthreads();
    for (int st = 128; st > 0; st >>= 1) {
      if (threadIdx.x < st) s[threadIdx.x] += s[threadIdx.x + st];
      __syncthreads();
    }
    // rank < nz_eff  ->  Z true -> adj 0 ; else A2 - A
    adj = (s[0] < nzeff) ? 0.0f : (a2 - a);
  }

  if (threadIdx.x == 0) {
    float val = th1[0] * w[e] + th2[0] * adj;
    out[e] = val > 0.0f ? val : 0.0f;
  }
}

// ---------------------------------------------------------------------------
// Launch
// ---------------------------------------------------------------------------
extern "C" void kernel_launch(void* const* d_in, const int* in_sizes, int n_in,
                              void* d_out, int out_size, void* d_ws,
                              size_t ws_size, hipStream_t stream) {
  const int* ei = (const int*)d_in[0];     // (2, E) int32
  const float* ew = (const float*)d_in[1]; // (E,)
  const float* X = (const float*)d_in[2];  // (N, F)
  const float* th1 = (const float*)d_in[3];
  const float* th2 = (const float*)d_in[4];
  float* out = (float*)d_out;

  // Workspace layout (G reuses A's slot after all consumers of A ran):
  const size_t MAT = (size_t)NN * NN * sizeof(float);  // 64 MB
  char* ws = (char*)d_ws;
  float* A = (float*)(ws);            // later: G
  float* A2 = (float*)(ws + MAT);
  float* sq = (float*)(ws + 2 * MAT);
  int* D = (int*)(ws + 2 * MAT + NN * 4);
  int* Dr = (int*)(ws + 2 * MAT + NN * 8);
  float* auv = (float*)(ws + 2 * MAT + NN * 12);
  float* a2uv = (float*)(ws + 2 * MAT + NN * 12 + EE * 4);
  float* thres = (float*)(ws + 2 * MAT + NN * 12 + EE * 8);

  const size_t nelems = (size_t)NN * NN;
  const int zblocks = (int)(nelems / 256);  // 65536

  // 1. A = clamp(scatter(w) + I, 1)
  k_zero<<<zblocks, 256, 0, stream>>>(A, nelems);
  k_scatter<<<EE / 256, 256, 0, stream>>>(ei, ew, A);
  k_finishA<<<zblocks, 256, 0, stream>>>(A);

  // 2. A2 = A @ A   (f32 WMMA)
  dim3 gg(NN / 64, NN / 128);
  k_gemm_nn<<<gg, 256, 0, stream>>>(A, A, A2, NN);

  // 3. diag zero, degrees, per-edge gathers (must precede G overwrite of A)
  k_zerodiag<<<NN / 256, 256, 0, stream>>>(A, A2);
  k_degrees<<<NN, 256, 0, stream>>>(A, A2, D, Dr);
  k_gather<<<EE / 256, 256, 0, stream>>>(ei, A, A2, auv, a2uv);

  // 4. G = X @ X^T (overwrites A slot), sq[u] = ||X_u||^2
  k_sq<<<NN / 256, 256, 0, stream>>>(X, sq);
  k_gemm_nt<<<gg, 256, 0, stream>>>(X, A);

  // 5. d_thres = mean(D) + 2*std(D)
  k_stats<<<1, 256, 0, stream>>>(D, thres);

  // 6. per-edge branch select + farther-neighbor count + output
  k_edges<<<EE, 256, 0, stream>>>(ei, ew, th1, th2, A2, /*G=*/A, sq, D, Dr,
                                  auv, a2uv, thres, out);
}